// HarmonicOscillator_36481452212630
// MI455X (gfx1250) — compile-verified
//
#include <hip/hip_runtime.h>

// ---------------------------------------------------------------------------
// HarmonicOscillator for MI455X (gfx1250, wave32).
// Store-bandwidth-bound: 122.9 MB out @ 23.3 TB/s => ~5.3us floor.
// Phase cumsum done hierarchically; intra-tile scan uses V_WMMA_F32_16X16X4_F32
// (P = L * X with lower-triangular ones), per ISA 7.12.2 operand layouts.
// ---------------------------------------------------------------------------

#define SRATE      24000.0f
#define T_TOTAL    240000
#define NFRAMES    500
#define BATCH      16
#define NH         8          // NUM_HARMONICS + 1 output channels
#define TILE       256
#define NT         938        // ceil(240000/256); last tile has 128 valid samples
#define NOISE_SC   0.03f
#define MIN_F      20.0f

typedef __attribute__((ext_vector_type(2))) float v2f;
typedef __attribute__((ext_vector_type(8))) float v8f;

__device__ __forceinline__ float lane_read(float v, int srcLane) {
    // wave32 gather: ds_bpermute uses byte index, bits [6:2]
    return __int_as_float(
        __builtin_amdgcn_ds_bpermute((srcLane & 31) << 2, __float_as_int(v)));
}

// torch F.interpolate(mode='linear', align_corners=False), scale=480
__device__ __forceinline__ float f0_interp(const float* __restrict__ row, int t) {
    float c = ((float)t + 0.5f) * (1.0f / 480.0f) - 0.5f;
    c = fmaxf(c, 0.0f);
    int i0 = (int)c;                       // c >= 0 so trunc == floor
    i0 = min(i0, NFRAMES - 1);
    int i1 = min(i0 + 1, NFRAMES - 1);
    float w = c - (float)i0;
    float a = row[i0];
    float b = row[i1];
    return fmaf(w, b - a, a);
}

// ---------------- kernel 1: per-tile phase sums (one wave per tile) --------
__global__ void __launch_bounds__(256) k_tile_sums(const float* __restrict__ f0,
                                                   float* __restrict__ tileSum) {
    int wid  = (blockIdx.x * blockDim.x + threadIdx.x) >> 5;
    int lane = threadIdx.x & 31;
    if (wid >= BATCH * NT) return;
    int b    = wid / NT;
    int tile = wid - b * NT;
    const float* row = f0 + b * NFRAMES;

    int t0 = tile * TILE + lane * 8;
    float s = 0.0f;
#pragma unroll
    for (int i = 0; i < 8; ++i) {
        int t = t0 + i;
        if (t < T_TOTAL) s += f0_interp(row, t) * (1.0f / SRATE);
    }
#pragma unroll
    for (int off = 16; off >= 1; off >>= 1)
        s += lane_read(s, lane ^ off);
    if (lane == 0) tileSum[wid] = s;
}

// ---------------- kernel 2: exclusive scan of tile sums (1 wave / batch) ---
__global__ void __launch_bounds__(32) k_scan_tiles(const float* __restrict__ tileSum,
                                                   float* __restrict__ tileOff) {
    int b    = blockIdx.x;
    int lane = threadIdx.x;
    const int per = (NT + 31) / 32;        // 30 tiles per lane
    int start = lane * per;

    float s = 0.0f;
    for (int i = 0; i < per; ++i) {
        int idx = start + i;
        if (idx < NT) s += tileSum[b * NT + idx];
    }
    float inc = s;
#pragma unroll
    for (int off = 1; off < 32; off <<= 1) {
        float y = lane_read(inc, lane - off);
        if (lane >= off) inc += y;
    }
    float run = inc - s;                   // exclusive prefix of lane chunks
    for (int i = 0; i < per; ++i) {
        int idx = start + i;
        if (idx < NT) {
            tileOff[b * NT + idx] = run;
            run += tileSum[b * NT + idx];
        }
    }
}

// ---------------- kernel 3: WMMA intra-tile scan + synthesis ---------------
__global__ void __launch_bounds__(256) k_osc(const float* __restrict__ f0,
                                             const float* __restrict__ tileOff,
                                             float* __restrict__ out) {
    int wid  = (blockIdx.x * blockDim.x + threadIdx.x) >> 5;
    int lane = threadIdx.x & 31;
    if (wid >= BATCH * NT) return;
    int b    = wid / NT;
    int tile = wid - b * NT;
    const float* row = f0 + b * NFRAMES;
    if (lane == 0) __builtin_prefetch(row, 0, 3);   // global_prefetch_b8

    int half     = lane >> 4;      // A/B K-split half, D row-half
    int cc       = lane & 15;      // column = time chunk index
    int tileBase = tile * TILE;

    // B operand samples: lane holds X(4kk+2*half+p, cc) = x[tileBase+16cc+4kk+2half+p]
    float xv[8];
#pragma unroll
    for (int kk = 0; kk < 4; ++kk) {
#pragma unroll
        for (int p = 0; p < 2; ++p) {
            int t = tileBase + 16 * cc + 4 * kk + 2 * half + p;
            float v = 0.0f;
            if (t < T_TOTAL) v = f0_interp(row, t) * (1.0f / SRATE);
            xv[kk * 2 + p] = v;
        }
    }

    // P = L * X  via 4x V_WMMA_F32_16X16X4_F32 (K chunks of 4, accumulate)
    v8f d = {};
#pragma unroll
    for (int kk = 0; kk < 4; ++kk) {
        int k0 = 4 * kk + 2 * half;        // this lane's two K columns of L
        int m  = cc;                       // this lane's A row
        v2f a;
        a.x = (k0     <= m) ? 1.0f : 0.0f;
        a.y = (k0 + 1 <= m) ? 1.0f : 0.0f;
        v2f bb;
        bb.x = xv[2 * kk];
        bb.y = xv[2 * kk + 1];
        d = __builtin_amdgcn_wmma_f32_16x16x4_f32(
                false, a, false, bb, (short)0, d, false, false);
    }

    // column totals P(15,c) live in d[7] on lanes 16..31; scan across 16 columns
    float tot = lane_read(d[7], 16 + cc);
    float inc = tot;
#pragma unroll
    for (int off = 1; off < 16; off <<= 1) {
        float y = lane_read(inc, lane - off);
        if (cc >= off) inc += y;
    }
    float base = (inc - tot) + tileOff[wid];   // exclusive col prefix + tile base

    float S[8];
#pragma unroll
    for (int j = 0; j < 8; ++j) S[j] = base + d[j];   // inclusive phase at t

    // this lane's output times: tl..tl+7 (rows j resp. j+8 of column cc)
    int tl = tileBase + 16 * cc + 8 * half;
    if (tl + 8 > T_TOTAL) return;              // tail tile, invalid columns

    float uv[8];
#pragma unroll
    for (int j = 0; j < 8; ++j)
        uv[j] = (f0_interp(row, tl + j) >= MIN_F) ? 1.0f : 0.0f;

#pragma unroll
    for (int h = 0; h < NH; ++h) {
        float hm = (float)(h + 1);
        float vals[8];
#pragma unroll
        for (int j = 0; j < 8; ++j) {
            float fr = __builtin_amdgcn_fractf(hm * S[j]);
            float sn = __builtin_amdgcn_sinf(fr);        // sin(2*pi*fr)
            // deterministic PCG hash -> Box-Muller noise
            unsigned idx = (unsigned)((b * NH + h) * T_TOTAL + tl + j);
            unsigned st  = idx * 747796405u + 2891336453u;
            unsigned w   = ((st >> ((st >> 28) + 4u)) ^ st) * 277803737u;
            w = (w >> 22) ^ w;
            float u1 = (float)((w & 0xFFFFu) + 1u) * (1.0f / 65536.0f);
            float u2 = (float)(w >> 16) * (1.0f / 65536.0f);
            float r  = __builtin_amdgcn_sqrtf(-1.3862944f * __builtin_amdgcn_logf(u1));
            float nz = r * __builtin_amdgcn_cosf(u2);    // cos(2*pi*u2)
            vals[j] = fmaf(nz, NOISE_SC, sn * uv[j]);
        }
        float* p = out + (size_t)(b * NH + h) * T_TOTAL + tl;   // 32B aligned
        reinterpret_cast<float4*>(p)[0] = make_float4(vals[0], vals[1], vals[2], vals[3]);
        reinterpret_cast<float4*>(p)[1] = make_float4(vals[4], vals[5], vals[6], vals[7]);
    }
}

// ---------------------------------------------------------------------------
extern "C" void kernel_launch(void* const* d_in, const int* in_sizes, int n_in,
                              void* d_out, int out_size, void* d_ws, size_t ws_size,
                              hipStream_t stream) {
    const float* f0 = (const float*)d_in[0];     // [16,1,500] fp32
    float* out      = (float*)d_out;             // [16,8,240000] fp32
    float* tileSum  = (float*)d_ws;              // 16*938 floats
    float* tileOff  = tileSum + BATCH * NT;      // 16*938 floats (120 KB total)

    int totalWaves = BATCH * NT;                 // 15008 tiles, 1 wave each
    int blocks     = (totalWaves * 32 + 255) / 256;   // 1876 WGs of 8 waves

    k_tile_sums<<<blocks, 256, 0, stream>>>(f0, tileSum);
    k_scan_tiles<<<BATCH, 32, 0, stream>>>(tileSum, tileOff);
    k_osc<<<blocks, 256, 0, stream>>>(f0, tileOff, out);
}